// CubEclayr_22247930593539
// MI455X (gfx1250) — compile-verified
//
#include <hip/hip_runtime.h>
#include <hip/hip_bf16.h>

// ---------------------------------------------------------------------------
// ECC (Euler characteristic curve) over a V-construction cubical complex.
//
// Plan (MI455X / gfx1250, wave32):
//   Kernel 1: per-plane signed histogram of cell filtration bins.
//             48 workgroups (one per (b,c) plane), 512 threads = 16 wave32s,
//             per-wave privatized LDS int histograms (ds_add atomics),
//             coalesced streaming loads of the 2x2 neighborhood.
//   Kernel 2: cumulative sum over the 32 bins expressed as
//             ECC = HIST(48x32) x U(32x32)  (U = upper-triangular ones)
//             using chained V_WMMA_F32_16X16X4_F32 (exact f32 integer math).
// ---------------------------------------------------------------------------

#define BB     16
#define CC     3
#define HH     224
#define WW     224
#define NIMG   (BB * CC)     // 48 planes
#define STEPS  32
#define NWAVE  16            // waves per block in kernel 1 (512 threads)

typedef __attribute__((ext_vector_type(2))) float v2f;
typedef __attribute__((ext_vector_type(8))) float v8f;

// smallest b in [0,32) with tseq[b] >= v, or 32 if none (== searchsorted 'left').
// Start from ceil(31*v) and correct against the actual f32 table to be robust
// to rounding at the bin boundaries.
__device__ __forceinline__ int bin_of(float v, const float* __restrict__ t) {
    int b = (int)ceilf(v * 31.0f);
    b = (b < 0) ? 0 : ((b > 31) ? 31 : b);
    while (t[b] < v) {                 // move right while threshold still < v
        if (++b == STEPS) return STEPS; // v above all thresholds -> masked out
    }
    while (b > 0 && t[b - 1] >= v) --b; // move left while previous also >= v
    return b;
}

// ---------------------------------------------------------------------------
// Kernel 1: signed histogram per plane. hist[img][k] (float) in workspace.
// ---------------------------------------------------------------------------
__global__ __launch_bounds__(NWAVE * 32)
void ecc_hist_kernel(const float* __restrict__ x, float* __restrict__ hist) {
    __shared__ float tseq[STEPS];
    __shared__ int   h[NWAVE][STEPS];   // per-wave privatized histograms

    const int tid  = threadIdx.x;
    const int wave = tid >> 5;

    if (tid < STEPS) tseq[tid] = (float)tid / 31.0f;
    for (int i = tid; i < NWAVE * STEPS; i += blockDim.x) ((int*)h)[i] = 0;
    __syncthreads();

    const float* __restrict__ img = x + (size_t)blockIdx.x * (HH * WW);
    int* __restrict__ myh = h[wave];

    for (int p = tid; p < HH * WW; p += blockDim.x) {
        const int i = p / WW;
        const int j = p - i * WW;
        const float v00 = img[p];

        // pixel cell (+1)
        {
            const int b = bin_of(v00, tseq);
            if (b < STEPS) atomicAdd(&myh[b], 1);
        }

        const bool hasR = (j < WW - 1);
        const bool hasD = (i < HH - 1);
        float v01 = 0.0f, v10 = 0.0f;

        // horizontal edge (-1)
        if (hasR) {
            v01 = img[p + 1];
            const int b = bin_of(fmaxf(v00, v01), tseq);
            if (b < STEPS) atomicAdd(&myh[b], -1);
        }
        // vertical edge (-1)
        if (hasD) {
            v10 = img[p + WW];
            const int b = bin_of(fmaxf(v00, v10), tseq);
            if (b < STEPS) atomicAdd(&myh[b], -1);
        }
        // unit square (+1)
        if (hasR && hasD) {
            const float v11 = img[p + WW + 1];
            const int b = bin_of(fmaxf(fmaxf(v00, v01), fmaxf(v10, v11)), tseq);
            if (b < STEPS) atomicAdd(&myh[b], 1);
        }
    }
    __syncthreads();

    if (tid < STEPS) {
        int s = 0;
        #pragma unroll
        for (int w = 0; w < NWAVE; ++w) s += h[w][tid];
        hist[blockIdx.x * STEPS + tid] = (float)s;
    }
}

// ---------------------------------------------------------------------------
// Kernel 2: ECC = HIST x U via V_WMMA_F32_16X16X4_F32.
//   6 waves: 3 row tiles (48 rows) x 2 col tiles (32 cols).
//   Per 16x16 tile: 8 chained WMMAs over K = 32 in slices of 4.
//
// A (16x4 f32) ISA layout: lane L, M = L%16;
//   VGPR0 -> K = (L<16 ? 0 : 2), VGPR1 -> K = (L<16 ? 1 : 3)   [CDNA5 ISA]
// B (4x16 f32) uses the mirrored striping: row K in the same position, N = L%16.
// D (16x16 f32, 8 VGPRs): VGPR v -> row M = v + (L<16 ? 0 : 8), N = L%16.
// ---------------------------------------------------------------------------
__global__ __launch_bounds__(192)
void ecc_cumsum_wmma(const float* __restrict__ hist, float* __restrict__ out) {
    const int wave = threadIdx.x >> 5;      // 0..5
    const int lane = threadIdx.x & 31;
    const int rt = wave >> 1;               // row tile 0..2
    const int ct = wave & 1;                // col tile 0..1

    const int l16  = lane & 15;
    const int hiK  = (lane < 16) ? 0 : 2;   // K pair selector for A/B slices
    const int rowg = rt * 16 + l16;         // A-matrix row (plane index)
    const int ng   = ct * 16 + l16;         // global output column (threshold)

    v8f acc = {};
    #pragma unroll
    for (int kk = 0; kk < 8; ++kk) {
        const int k0 = kk * 4 + hiK;        // this lane's K indices: k0, k0+1
        v2f a, b;
        a.x = hist[rowg * STEPS + k0];
        a.y = hist[rowg * STEPS + k0 + 1];
        // U[k][n] = (k <= n) ? 1 : 0  (upper-triangular ones -> prefix sum)
        b.x = (k0     <= ng) ? 1.0f : 0.0f;
        b.y = (k0 + 1 <= ng) ? 1.0f : 0.0f;
        acc = __builtin_amdgcn_wmma_f32_16x16x4_f32(
                  /*neg_a=*/false, a, /*neg_b=*/false, b,
                  /*c_mod=*/(short)0, acc, /*reuse_a=*/false, /*reuse_b=*/false);
    }

    // scatter D per the 16x16 f32 C/D layout
    const int rbase = rt * 16 + ((lane < 16) ? 0 : 8);
    #pragma unroll
    for (int v = 0; v < 8; ++v) {
        out[(rbase + v) * STEPS + ng] = acc[v];
    }
}

// ---------------------------------------------------------------------------
extern "C" void kernel_launch(void* const* d_in, const int* in_sizes, int n_in,
                              void* d_out, int out_size, void* d_ws, size_t ws_size,
                              hipStream_t stream) {
    const float* x   = (const float*)d_in[0];      // [16,3,224,224] f32
    float*       out = (float*)d_out;              // [16, 96] == flat [48][32] f32
    float*       hist = (float*)d_ws;              // [48][32] f32 scratch (6 KB)

    ecc_hist_kernel<<<NIMG, NWAVE * 32, 0, stream>>>(x, hist);
    ecc_cumsum_wmma<<<1, 192, 0, stream>>>(hist, out);
}